// MeanOthers_23021024707408
// MI455X (gfx1250) — compile-verified
//
#include <hip/hip_runtime.h>
#include <stdint.h>

// Leave-one-out mean along dim 1: out[b,i] = (row_sum[b] - x[b,i]) / (N-1)
// x: 8192 x 16384 f32. Pure streaming problem (1 GiB HBM traffic, ~0.27 GFLOP)
// -> HBM-bound, floor ~46us @ 23.3 TB/s. Strategy: one workgroup per row,
// stage the 64 KiB row into LDS ONCE via gfx1250 async LDS-DMA loads
// (ASYNCcnt-tracked), reduce in-wave with wave32 shuffles, write back with
// non-temporal stores.

#define N_COLS   16384
#define BLOCK    512
#define NWAVES   (BLOCK / 32)            // 16 wave32 waves
#define CHUNKS   (N_COLS / (BLOCK * 4))  // 8 float4 chunks per thread

// Native clang vector types (HIP's float4 is a class; builtins need these).
typedef float v4f __attribute__((ext_vector_type(4)));
typedef int   v4i __attribute__((ext_vector_type(4)));
typedef __attribute__((address_space(1))) v4i gbl_v4i;   // global pointer payload type
typedef __attribute__((address_space(3))) v4i lds_v4i;   // LDS pointer payload type

#if __has_builtin(__builtin_amdgcn_global_load_async_to_lds_b128) && \
    __has_builtin(__builtin_amdgcn_s_wait_asynccnt)
#define USE_ASYNC_LDS 1
#else
#define USE_ASYNC_LDS 0
#endif

// Generic LDS pointers carry the LDS byte offset in their low 32 bits
// (ISA 10.2); truncating through u32 yields a valid
// 32-bit addrspace(3) pointer. Global pointers pass through unchanged.
__device__ __forceinline__ lds_v4i* as_lds(void* p) {
    return (lds_v4i*)(uint32_t)(uintptr_t)p;
}
__device__ __forceinline__ gbl_v4i* as_global(const void* p) {
    return (gbl_v4i*)(uintptr_t)p;
}

__global__ __launch_bounds__(BLOCK)
void MeanOthers_23021024707408_kernel(const float* __restrict__ x,
                                      float* __restrict__ out,
                                      int n_rows) {
    __shared__ float tile[N_COLS];  // 64 KiB staging buffer (reused for wave sums)

    const int row = blockIdx.x;
    if (row >= n_rows) return;

    const float* xrow = x   + (size_t)row * N_COLS;
    float*       orow = out + (size_t)row * N_COLS;

    const int t    = threadIdx.x;
    const int wave = t >> 5;   // wave32 on gfx1250
    const int lane = t & 31;

    v4f v[CHUNKS];

#if USE_ASYNC_LDS
    // ---- Phase 1: async DMA the row into LDS (tracked by ASYNCcnt) ----
    // Lane-private regions: chunk index c = t + BLOCK*k, 16 bytes per chunk.
#pragma unroll
    for (int k = 0; k < CHUNKS; ++k) {
        const int c = t + BLOCK * k;
        __builtin_amdgcn_global_load_async_to_lds_b128(
            as_global(xrow + 4 * c), as_lds(&tile[4 * c]),
            /*offset=*/0, /*cpol=*/0);
    }
    // Each wave reads back only LDS it wrote itself -> waiting on our own
    // ASYNCcnt is sufficient, no workgroup barrier needed yet.
    __builtin_amdgcn_s_wait_asynccnt(0);
#pragma unroll
    for (int k = 0; k < CHUNKS; ++k) {
        const int c = t + BLOCK * k;
        v[k] = *(const v4f*)&tile[4 * c];
    }
#else
    // Fallback: straight vectorized non-temporal global loads.
#pragma unroll
    for (int k = 0; k < CHUNKS; ++k) {
        const int c = t + BLOCK * k;
        v[k] = __builtin_nontemporal_load((const v4f*)(xrow + 4 * c));
    }
#endif

    // ---- Phase 2: row sum ----
    float s = 0.0f;
#pragma unroll
    for (int k = 0; k < CHUNKS; ++k)
        s += (v[k].x + v[k].y) + (v[k].z + v[k].w);

    // wave32 butterfly reduction
#pragma unroll
    for (int off = 16; off >= 1; off >>= 1)
        s += __shfl_xor(s, off, 32);

    // Exchange the 16 wave sums through LDS. Tile data is already in
    // registers, so tile[0..15] can be reused after the barrier.
    __syncthreads();
    if (lane == 0) tile[wave] = s;
    __syncthreads();

    float row_sum = 0.0f;
#pragma unroll
    for (int w = 0; w < NWAVES; ++w) row_sum += tile[w];  // LDS broadcast reads

    // ---- Phase 3: out = (row_sum - x) * inv, write-once -> NT stores ----
    const float inv  = 1.0f / (float)(N_COLS - 1);
    const float base = row_sum * inv;
    const float ninv = -inv;
#pragma unroll
    for (int k = 0; k < CHUNKS; ++k) {
        const int c = t + BLOCK * k;
        v4f r;
        r.x = __builtin_fmaf(v[k].x, ninv, base);
        r.y = __builtin_fmaf(v[k].y, ninv, base);
        r.z = __builtin_fmaf(v[k].z, ninv, base);
        r.w = __builtin_fmaf(v[k].w, ninv, base);
        __builtin_nontemporal_store(r, (v4f*)(orow + 4 * c));
    }
}

extern "C" void kernel_launch(void* const* d_in, const int* in_sizes, int n_in,
                              void* d_out, int out_size, void* d_ws, size_t ws_size,
                              hipStream_t stream) {
    const float* x = (const float*)d_in[0];
    float* out = (float*)d_out;
    const int n_rows = in_sizes[0] / N_COLS;  // 8192 for the reference shapes
    MeanOthers_23021024707408_kernel<<<dim3(n_rows), dim3(BLOCK), 0, stream>>>(
        x, out, n_rows);
}